// GAT_69587060130269
// MI455X (gfx1250) — compile-verified
//
#include <hip/hip_runtime.h>

// ---------------------------------------------------------------------------
// GAT layer for MI455X (gfx1250, wave32).
//   g = h @ W1            -> f32 WMMA 16x16x4 (W transposed+swizzled in LDS)
//   s_i/s_j               -> fused into GEMM epilogue (shfl_xor reduce)
//   segment softmax       -> HW float atomic max/add (L2-resident)
//   h'[src] += a*g[dst]   -> wave-per-edge row gather + global_atomic_add_f32
//   out = relu(h' @ W2)   -> f32 WMMA 16x16x4
// ---------------------------------------------------------------------------

typedef float v2f __attribute__((ext_vector_type(2)));
typedef float v8f __attribute__((ext_vector_type(8)));

#define FEAT  128
#define NBATCH 2

// float atomic max that lowers to HW int atomics (works for all sign mixes,
// buffer must be initialized to -inf):
__device__ __forceinline__ void atomicMaxF(float* addr, float val) {
  if (val >= 0.0f) {
    atomicMax((int*)addr, __float_as_int(val));
  } else {
    atomicMin((unsigned int*)addr, __float_as_uint(val));
  }
}

__global__ void fill_kernel(float* __restrict__ p, long long n, float v) {
  long long i = (long long)blockIdx.x * blockDim.x + threadIdx.x;
  if (i < n) p[i] = v;
}

// ---------------------------------------------------------------------------
// GEMM: Y[nrows x 128] = X[nrows x 128] * W[128 x 128]
// Block = 256 threads = 8 waves; each wave computes a 16-row x 128-col strip.
//
// W is staged TRANSPOSED in LDS with a per-column K-rotation swizzle:
//   element (k, col) -> Wt[col*128 + ((k + 4*(col&15)) & 127)]
// so a B fragment (W[kb][col], W[kb+1][col]) is one aligned ds_load_b64 and
// the 32 lanes of a wave hit 64 distinct LDS banks (conflict-free).
//
// A fragment (V_WMMA_F32_16X16X4_F32, 16x4):
//   lane M = lane&15 ; VGPR0/1 hold K = k + 2*(lane>>4) + {0,1}
// B fragment (4x16): lane N = lane&15 ; VGPR0/1 hold rows K = k+2*(lane>>4)+{0,1}
// C/D (16x16): 8 VGPRs, M = vgpr + 8*(lane>>4), N = lane&15
// ---------------------------------------------------------------------------
template <bool RELU, bool FUSE>
__global__ __launch_bounds__(256)
void gemm128_wmma(const float* __restrict__ X, const float* __restrict__ W,
                  float* __restrict__ Y, const float* __restrict__ avec,
                  float* __restrict__ s_i, float* __restrict__ s_j, int nrows) {
  __shared__ float Wt[FEAT * FEAT];   // 64 KB, transposed + swizzled

  const int tid = threadIdx.x;
  for (int i = tid; i < FEAT * FEAT; i += 256) {
    const int k   = i >> 7;     // source row of W
    const int col = i & 127;    // source col of W
    Wt[col * FEAT + ((k + 4 * (col & 15)) & 127)] = W[i];
  }
  __syncthreads();

  const int wave = tid >> 5;
  const int lane = tid & 31;
  const int half = lane >> 4;   // 0 or 1
  const int l15  = lane & 15;
  const int row0 = blockIdx.x * 128 + wave * 16;

  v8f acc[8];
#pragma unroll
  for (int t = 0; t < 8; ++t) acc[t] = (v8f)0.0f;

  int arow = row0 + l15;
  if (arow >= nrows) arow = nrows - 1;       // keep EXEC all-ones for WMMA
  const float* __restrict__ Xr = X + (long long)arow * FEAT;

  const int sw = 4 * l15;                    // per-lane K rotation
  for (int k = 0; k < FEAT; k += 4) {
    const int kb = k + half * 2;
    v2f afrag;
    afrag.x = Xr[kb];
    afrag.y = Xr[kb + 1];
#pragma unroll
    for (int t = 0; t < 8; ++t) {
      const int col = t * 16 + l15;
      const v2f bfrag =
          *(const v2f*)&Wt[col * FEAT + ((kb + sw) & 127)];  // ds_load_b64
      acc[t] = __builtin_amdgcn_wmma_f32_16x16x4_f32(
          false, afrag, false, bfrag, (short)0, acc[t], false, false);
    }
  }

  // --- store D tiles ---
#pragma unroll
  for (int r = 0; r < 8; ++r) {
    const int row = row0 + r + 8 * half;
    if (row < nrows) {
#pragma unroll
      for (int t = 0; t < 8; ++t) {
        float v = acc[t][r];
        if (RELU) v = fmaxf(v, 0.0f);
        Y[(long long)row * FEAT + t * 16 + l15] = v;
      }
    }
  }

  // --- fused s_i / s_j (per-row dot with attention vector halves) ---
  if (FUSE) {
    float aI[8], aJ[8];
#pragma unroll
    for (int t = 0; t < 8; ++t) {
      aI[t] = avec[t * 16 + l15];
      aJ[t] = avec[FEAT + t * 16 + l15];
    }
#pragma unroll
    for (int r = 0; r < 8; ++r) {
      float si = 0.0f, sj = 0.0f;
#pragma unroll
      for (int t = 0; t < 8; ++t) {
        si += acc[t][r] * aI[t];
        sj += acc[t][r] * aJ[t];
      }
      // reduce across the 16 lanes holding this row (offsets stay in-half)
#pragma unroll
      for (int off = 8; off > 0; off >>= 1) {
        si += __shfl_xor(si, off, 32);
        sj += __shfl_xor(sj, off, 32);
      }
      if (l15 == 0) {
        const int row = row0 + r + 8 * half;
        if (row < nrows) { s_i[row] = si; s_j[row] = sj; }
      }
    }
  }
}

// ---------------------------------------------------------------------------
// Edge phase 1: segment max of leaky-relu scores, grouped by src.
// ---------------------------------------------------------------------------
__global__ void edge_max_kernel(const int* __restrict__ src, const int* __restrict__ dst,
                                const float* __restrict__ s_i, const float* __restrict__ s_j,
                                float* __restrict__ m, int E, int N) {
  const int e = blockIdx.x * 256 + threadIdx.x;
  if (e >= E) return;
  const int s = src[e], d = dst[e];
#pragma unroll
  for (int b = 0; b < NBATCH; ++b) {
    float x = s_i[b * N + s] + s_j[b * N + d];
    x = x > 0.0f ? x : 0.2f * x;
    atomicMaxF(&m[b * N + s], x);
  }
}

// ---------------------------------------------------------------------------
// Edge phase 2: ex = exp(e - m[src]); denom[src] += ex
// ---------------------------------------------------------------------------
__global__ void edge_exp_kernel(const int* __restrict__ src, const int* __restrict__ dst,
                                const float* __restrict__ s_i, const float* __restrict__ s_j,
                                const float* __restrict__ m, float* __restrict__ ex,
                                float* __restrict__ denom, int E, int N) {
  const int e = blockIdx.x * 256 + threadIdx.x;
  if (e >= E) return;
  const int s = src[e], d = dst[e];
#pragma unroll
  for (int b = 0; b < NBATCH; ++b) {
    float x = s_i[b * N + s] + s_j[b * N + d];
    x = x > 0.0f ? x : 0.2f * x;
    const float v = __expf(x - m[b * N + s]);
    ex[(long long)b * E + e] = v;
    atomicAdd(&denom[b * N + s], v);
  }
}

// ---------------------------------------------------------------------------
// Edge phase 3: h'[b, src] += alpha * g[b, dst].  One wave per (edge, batch):
// 32 lanes x float4 = one full 128-float row, coalesced 512B gather, then
// non-returning global f32 atomic adds.
// ---------------------------------------------------------------------------
__global__ void scatter_kernel(const int* __restrict__ src, const int* __restrict__ dst,
                               const float* __restrict__ g, const float* __restrict__ ex,
                               const float* __restrict__ denom, float* __restrict__ hp,
                               int E, int N) {
  const int gid  = blockIdx.x * 256 + threadIdx.x;
  const int w    = gid >> 5;
  const int lane = gid & 31;
  const int e    = w >> 1;
  const int b    = w & 1;
  if (e >= E) return;

  const int s = src[e], d = dst[e];
  const float alpha = ex[(long long)b * E + e] / denom[(long long)b * N + s];

  const float4* __restrict__ gp =
      (const float4*)(g + ((long long)b * N + d) * FEAT);
  const float4 v = gp[lane];

  float* __restrict__ o = hp + ((long long)b * N + s) * FEAT + lane * 4;
  atomicAdd(o + 0, v.x * alpha);
  atomicAdd(o + 1, v.y * alpha);
  atomicAdd(o + 2, v.z * alpha);
  atomicAdd(o + 3, v.w * alpha);
}

// ---------------------------------------------------------------------------
extern "C" void kernel_launch(void* const* d_in, const int* in_sizes, int n_in,
                              void* d_out, int out_size, void* d_ws, size_t ws_size,
                              hipStream_t stream) {
  const float* h  = (const float*)d_in[0];   // [B,N,F]
  const float* W1 = (const float*)d_in[1];   // [F,F]
  const float* W2 = (const float*)d_in[2];   // [F,OUT]
  const float* a  = (const float*)d_in[3];   // [2F,1]
  const int*  src = (const int*)d_in[4];     // [E]
  const int*  dst = (const int*)d_in[5];     // [E]
  float* out = (float*)d_out;

  const int N = in_sizes[0] / (NBATCH * FEAT);   // 50000
  const int E = in_sizes[4];                     // 800000
  const long long BN    = (long long)NBATCH * N; // rows of the fused (b,n) space
  const long long rowsF = BN * FEAT;

  // workspace layout (floats)
  float* ws    = (float*)d_ws;
  float* g     = ws;                    // [B*N, F]
  float* s_i   = g + rowsF;             // [B*N]
  float* s_j   = s_i + BN;              // [B*N]
  float* m     = s_j + BN;              // [B*N]
  float* denom = m + BN;                // [B*N]
  float* ex    = denom + BN;            // [B*E]
  float* hp    = ex + (long long)NBATCH * E;   // [B*N, F]

  const int gemmBlocks = (int)((BN + 127) / 128);
  const int edgeBlocks = (E + 255) / 256;

  // init: m = -inf, denom = 0, h' = 0
  fill_kernel<<<(int)((BN + 255) / 256), 256, 0, stream>>>(m, BN, -__builtin_inff());
  fill_kernel<<<(int)((BN + 255) / 256), 256, 0, stream>>>(denom, BN, 0.0f);
  fill_kernel<<<(int)((rowsF + 255) / 256), 256, 0, stream>>>(hp, rowsF, 0.0f);

  // g = h @ W1, fused s_i/s_j
  gemm128_wmma<false, true><<<gemmBlocks, 256, 0, stream>>>(
      h, W1, g, a, s_i, s_j, (int)BN);

  // segment softmax over edges grouped by src
  edge_max_kernel<<<edgeBlocks, 256, 0, stream>>>(src, dst, s_i, s_j, m, E, N);
  edge_exp_kernel<<<edgeBlocks, 256, 0, stream>>>(src, dst, s_i, s_j, m, ex, denom, E, N);

  // h'[src] += alpha * g[dst]   (one wave per edge per batch)
  const long long waves = (long long)NBATCH * E;
  const int scatterBlocks = (int)((waves * 32 + 255) / 256);
  scatter_kernel<<<scatterBlocks, 256, 0, stream>>>(src, dst, g, ex, denom, hp, E, N);

  // out = relu(h' @ W2)
  gemm128_wmma<true, false><<<gemmBlocks, 256, 0, stream>>>(
      hp, W2, out, nullptr, nullptr, nullptr, (int)BN);
}